// SingleHeadAttention_80083960201725
// MI455X (gfx1250) — compile-verified
//
#include <hip/hip_runtime.h>
#include <hip/hip_bf16.h>

// ---- problem dims (from reference) ----
#define B_  4
#define S_  2048
#define DM  1024
#define DK  64

typedef __attribute__((ext_vector_type(2))) float v2f;
typedef __attribute__((ext_vector_type(8))) float v8f;

#define LSTRIDE 68      // padded LDS row stride in floats: 16B-aligned rows, conflict-free b64 column walks
#define PSTRIDE 18      // padded stride for 16x16 P scratch

// f32 WMMA: D(16x16) = A(16x4) * B(4x16) + C.  8-arg VOP3P form.
__device__ __forceinline__ v8f wmma_f32(v2f a, v2f b, v8f c) {
    return __builtin_amdgcn_wmma_f32_16x16x4_f32(false, a, false, b, (short)0, c, false, false);
}

// ---- CDNA5 async global->LDS (ASYNCcnt path), with sync fallback ----
#if __has_builtin(__builtin_amdgcn_global_load_async_to_lds_b128)
#define HAS_ASYNC_LDS 1
#else
#define HAS_ASYNC_LDS 0
#endif

typedef int i4v __attribute__((vector_size(16)));                 // b128 payload
typedef __attribute__((address_space(1))) i4v* g4p;               // global src
typedef __attribute__((address_space(3))) i4v* l4p;               // LDS dst

__device__ __forceinline__ void async_copy16(const float* g, float* l) {
#if HAS_ASYNC_LDS
    __builtin_amdgcn_global_load_async_to_lds_b128(
        (g4p)(size_t)g, (l4p)(size_t)l, 0, 0);
#else
    *(float4*)l = *(const float4*)g;
#endif
}

__device__ __forceinline__ void wait_async() {
#if HAS_ASYNC_LDS
  #if __has_builtin(__builtin_amdgcn_s_wait_asynccnt)
    __builtin_amdgcn_s_wait_asynccnt(0);
  #else
    asm volatile("s_wait_asynccnt 0" ::: "memory");
  #endif
#endif
}

// ---------------------------------------------------------------------------
// Kernel 1: row-wise projection  P[M,64] = X[M,1024] @ W[1024,64]
// grid.x = M/128, grid.y selects (Q,WQ)->(q) / (K,WK)->(k) / (V,WV)->(v)
// ---------------------------------------------------------------------------
__global__ __launch_bounds__(256)
void proj_kernel(const float* __restrict__ Q, const float* __restrict__ K,
                 const float* __restrict__ V,
                 const float* __restrict__ WQ, const float* __restrict__ WK,
                 const float* __restrict__ WV,
                 float* __restrict__ qp, float* __restrict__ kp,
                 float* __restrict__ vp) {
    __shared__ float Xs[128 * LSTRIDE];       // X tile, natural [row][k]
    __shared__ float Wt[64 * LSTRIDE];        // W chunk, transposed [n][k]

    const float* X = (blockIdx.y == 0) ? Q : (blockIdx.y == 1) ? K : V;
    const float* W = (blockIdx.y == 0) ? WQ : (blockIdx.y == 1) ? WK : WV;
    float*       P = (blockIdx.y == 0) ? qp : (blockIdx.y == 1) ? kp : vp;

    const int tid  = threadIdx.x;
    const int lane = tid & 31;
    const int wave = tid >> 5;
    const int hx   = lane >> 4;    // which 16-lane half
    const int l16  = lane & 15;
    const int row0 = blockIdx.x * 128;
    const int wrow = row0 + wave * 16;

    v8f acc[4] = {};

    for (int c0 = 0; c0 < DM; c0 += 64) {
        __syncthreads();
        // stage X tile 128x64: async global->LDS, 16B per op
        for (int i = tid; i < 128 * 16; i += 256) {
            int r = i >> 4, cc = (i & 15) * 4;
            async_copy16(&X[(size_t)(row0 + r) * DM + c0 + cc],
                         &Xs[r * LSTRIDE + cc]);
        }
        // stage W chunk 64x64 transposed: Wt[n][k]
        for (int i = tid; i < 64 * 16; i += 256) {
            int r = i >> 4, cc = (i & 15) * 4;     // r = k index, cc = n base
            float4 x = *(const float4*)&W[(size_t)(c0 + r) * DK + cc];
            Wt[(cc + 0) * LSTRIDE + r] = x.x;
            Wt[(cc + 1) * LSTRIDE + r] = x.y;
            Wt[(cc + 2) * LSTRIDE + r] = x.z;
            Wt[(cc + 3) * LSTRIDE + r] = x.w;
        }
        wait_async();
        __syncthreads();

        #pragma unroll
        for (int t = 0; t < 16; ++t) {
            const int c = 4 * t + 2 * hx;
            // A frag: lane l16 = row M, K = c, c+1  (contiguous b64)
            v2f a = *(const v2f*)&Xs[(wave * 16 + l16) * LSTRIDE + c];
            #pragma unroll
            for (int n = 0; n < 4; ++n) {
                // B frag: lane l16 = column N; K = c, c+1 contiguous in Wt
                v2f b = *(const v2f*)&Wt[(n * 16 + l16) * LSTRIDE + c];
                acc[n] = wmma_f32(a, b, acc[n]);
            }
        }
    }

    // C layout: VGPR r -> row (r + 8*hx), lane l16 -> col
    #pragma unroll
    for (int n = 0; n < 4; ++n)
        #pragma unroll
        for (int r = 0; r < 8; ++r)
            P[(size_t)(wrow + r + 8 * hx) * DK + n * 16 + l16] = acc[n][r];
}

// ---------------------------------------------------------------------------
// Kernel 2: causal flash attention over projected q,k,v  ([B,S,64] f32)
// one block per (batch, 128 q rows); 8 waves, 16 q rows per wave
// ---------------------------------------------------------------------------
__global__ __launch_bounds__(256)
void attn_kernel(const float* __restrict__ q, const float* __restrict__ k,
                 const float* __restrict__ v, float* __restrict__ H) {
    __shared__ float Ks[64 * LSTRIDE];        // K chunk, natural [kv][dk]
    __shared__ float Vt[64 * LSTRIDE];        // V chunk, transposed [vcol][kv]
    __shared__ float Ps[8][16 * PSTRIDE];     // per-wave P-tile transpose scratch

    const int tid  = threadIdx.x;
    const int lane = tid & 31;
    const int wave = tid >> 5;
    const int hx   = lane >> 4;
    const int l16  = lane & 15;
    const int b    = blockIdx.x >> 4;         // S/128 = 16 q blocks per batch
    const int qb   = blockIdx.x & 15;

    const float* qg = q + (size_t)b * S_ * DK;
    const float* kg = k + (size_t)b * S_ * DK;
    const float* vg = v + (size_t)b * S_ * DK;
    const int qr0 = qb * 128 + wave * 16;     // first q row of this wave

    // q A-fragments in registers, pre-scaled by 1/sqrt(64)
    v2f qa[16];
    #pragma unroll
    for (int t = 0; t < 16; ++t) {
        const float* p = qg + (size_t)(qr0 + l16) * DK + 4 * t + 2 * hx;
        qa[t][0] = p[0] * 0.125f;
        qa[t][1] = p[1] * 0.125f;
    }

    v8f acc[4] = {};
    float mrow[8], lrow[8];
    #pragma unroll
    for (int r = 0; r < 8; ++r) { mrow[r] = -1e30f; lrow[r] = 0.f; }

    const int nchunks = 2 * qb + 2;           // causal: kv chunks of 64 rows
    for (int j = 0; j < nchunks; ++j) {
        const int kv0 = j * 64;
        __syncthreads();
        // K chunk: async global->LDS (natural layout)
        for (int i = tid; i < 64 * 16; i += 256) {
            int r = i >> 4, cc = (i & 15) * 4;
            async_copy16(&kg[(size_t)(kv0 + r) * DK + cc],
                         &Ks[r * LSTRIDE + cc]);
        }
        // V chunk: staged transposed Vt[vcol][kv]
        for (int i = tid; i < 64 * 16; i += 256) {
            int r = i >> 4, cc = (i & 15) * 4;
            float4 x = *(const float4*)&vg[(size_t)(kv0 + r) * DK + cc];
            Vt[(cc + 0) * LSTRIDE + r] = x.x;
            Vt[(cc + 1) * LSTRIDE + r] = x.y;
            Vt[(cc + 2) * LSTRIDE + r] = x.z;
            Vt[(cc + 3) * LSTRIDE + r] = x.w;
        }
        // prefetch next causal chunk while this one computes
        if (j + 1 < nchunks) {
            const size_t nr = (size_t)(kv0 + 64 + (tid & 63)) * DK;
            __builtin_prefetch(&kg[nr], 0, 0);
            __builtin_prefetch(&vg[nr], 0, 0);
        }
        wait_async();
        __syncthreads();

        // ---- scores: 4 subtiles of 16 kv columns, inner dim d_k = 64 ----
        v8f st[4] = {};
        #pragma unroll
        for (int t = 0; t < 16; ++t) {
            const int c = 4 * t + 2 * hx;
            #pragma unroll
            for (int s = 0; s < 4; ++s) {
                // B = K^T: column N = kv row (s*16+l16); K rows c, c+1 contiguous
                v2f bf = *(const v2f*)&Ks[(s * 16 + l16) * LSTRIDE + c];
                st[s] = wmma_f32(qa[t], bf, st[s]);
            }
        }

        // ---- causal mask + online softmax (per row = VGPR r + half) ----
        float mnew[8];
        #pragma unroll
        for (int r = 0; r < 8; ++r) {
            const int row = qr0 + r + 8 * hx;
            float mx = mrow[r];
            #pragma unroll
            for (int s = 0; s < 4; ++s) {
                const int col = kv0 + s * 16 + l16;
                float val = (col <= row) ? st[s][r] : -1e30f;
                st[s][r] = val;
                mx = fmaxf(mx, val);
            }
            #pragma unroll
            for (int o = 1; o < 16; o <<= 1)       // stays within 16-lane half
                mx = fmaxf(mx, __shfl_xor(mx, o, 32));
            mnew[r] = mx;
        }
        #pragma unroll
        for (int r = 0; r < 8; ++r) {
            const float fr = __expf(mrow[r] - mnew[r]);
            float sum = 0.f;
            #pragma unroll
            for (int s = 0; s < 4; ++s) {
                float pv = __expf(st[s][r] - mnew[r]);
                st[s][r] = pv;
                sum += pv;
            }
            #pragma unroll
            for (int o = 1; o < 16; o <<= 1)
                sum += __shfl_xor(sum, o, 32);
            lrow[r] = lrow[r] * fr + sum;
            mrow[r] = mnew[r];
            #pragma unroll
            for (int n = 0; n < 4; ++n) acc[n][r] *= fr;
        }

        // ---- attn @ V: stage each 16x16 P tile via LDS (C-layout -> A-layout) ----
        #pragma unroll
        for (int s = 0; s < 4; ++s) {
            #pragma unroll
            for (int r = 0; r < 8; ++r)
                Ps[wave][(r + 8 * hx) * PSTRIDE + l16] = st[s][r];
            __syncthreads();   // block-uniform; orders per-wave LDS RAW safely
            #pragma unroll
            for (int tt = 0; tt < 4; ++tt) {
                const int c2 = 4 * tt + 2 * hx;
                // A frag of P: row M = l16, K = c2, c2+1 (contiguous b64)
                v2f a = *(const v2f*)&Ps[wave][l16 * PSTRIDE + c2];
                const int cv = s * 16 + 4 * tt + 2 * hx;   // kv row in chunk (even)
                #pragma unroll
                for (int n = 0; n < 4; ++n) {
                    // B = V: rows = kv, cols = v-dim; contiguous in Vt[vcol][kv]
                    v2f bf = *(const v2f*)&Vt[(n * 16 + l16) * LSTRIDE + cv];
                    acc[n] = wmma_f32(a, bf, acc[n]);
                }
            }
            __syncthreads();
        }
    }

    // ---- epilogue: H = acc / l ----
    float* Hg = H + (size_t)b * S_ * DK;
    #pragma unroll
    for (int r = 0; r < 8; ++r) {
        const float invl = 1.f / lrow[r];
        #pragma unroll
        for (int n = 0; n < 4; ++n)
            Hg[(size_t)(qr0 + r + 8 * hx) * DK + n * 16 + l16] = acc[n][r] * invl;
    }
}

// ---------------------------------------------------------------------------
extern "C" void kernel_launch(void* const* d_in, const int* in_sizes, int n_in,
                              void* d_out, int out_size, void* d_ws, size_t ws_size,
                              hipStream_t stream) {
    const float* Q  = (const float*)d_in[0];
    const float* K  = (const float*)d_in[1];
    const float* V  = (const float*)d_in[2];
    const float* WQ = (const float*)d_in[3];
    const float* WK = (const float*)d_in[4];
    const float* WV = (const float*)d_in[5];
    // d_in[6] = tril mask: implemented analytically (col <= row), not read.

    float* out = (float*)d_out;
    float* qp  = (float*)d_ws;                       // [B*S, 64] each
    float* kp  = qp + (size_t)B_ * S_ * DK;
    float* vp  = kp + (size_t)B_ * S_ * DK;

    dim3 pgrid(B_ * S_ / 128, 3);
    proj_kernel<<<pgrid, 256, 0, stream>>>(Q, K, V, WQ, WK, WV, qp, kp, vp);

    attn_kernel<<<dim3(B_ * S_ / 128), 256, 0, stream>>>(qp, kp, vp, out);
}